// cAttentionNoValue_16518444221234
// MI455X (gfx1250) — compile-verified
//
#include <hip/hip_runtime.h>

// Problem constants (from reference): B=4, N=4096, ID_EMBED_DIM=QK_DIM=128, E=127
#define BATCH 4
#define NROW  4096
#define S1    32            // split-K factor for the t-reduction
#define R1    (NROW / S1)   // 128 rows per split block
#define W3    4             // waves per block in the fx pass

typedef __attribute__((ext_vector_type(2))) float v2f;
typedef __attribute__((ext_vector_type(8))) float v8f;

#if defined(__has_builtin)
#if __has_builtin(__builtin_amdgcn_global_load_async_to_lds_b32)
#define HAVE_ASYNC_LDS 1
#endif
#endif

// Address-space-qualified pointer types for the async global->LDS builtin
// (signature per hipcc diagnostic: (addrspace(1) int*, addrspace(3) int*, i32, i32))
typedef __attribute__((address_space(1))) int gas_int;
typedef __attribute__((address_space(3))) int las_int;

// Workspace layout (in floats)
#define WS_PARTIAL 0                         // BATCH*S1*128 split-K partials of t
#define WS_SVP     (BATCH * S1 * 128)        // BATCH*S1 partials of sum(val)
#define WS_U       (WS_SVP + BATCH * S1)     // BATCH*128 : u[b] = Wq^T s[b]
#define WS_C       (WS_U + BATCH * 128)      // BATCH     : c[b] = bq . s[b]

// ---------------------------------------------------------------------------
// Kernel 1: split-K reduction  t_partial[b,split,d] = sum_j h0[b,j,d]*val[b,j]
// h0[:, :, 0] = val, h0[:, :, 1:128] = id_embed. 128 threads = one d each.
// Streaming, fully coalesced over d; val[b,j] is block-uniform (scalar load).
// ---------------------------------------------------------------------------
__global__ __launch_bounds__(128) void k1_reduce(const float* __restrict__ val,
                                                 const float* __restrict__ id_embed,
                                                 float* __restrict__ ws) {
  const int d     = threadIdx.x;        // 0..127 (h0 feature)
  const int blk   = blockIdx.x;         // b*S1 + split
  const int base  = (blk / S1) * NROW + (blk % S1) * R1;
  float acc = 0.f, accsv = 0.f;
  for (int jj = 0; jj < R1; ++jj) {
    const int gi = base + jj;           // flat (b, j) row index
    const float v = val[gi];
    const float x = (d == 0) ? v : id_embed[(size_t)gi * 127 + (d - 1)];
    acc = fmaf(x, v, acc);
    accsv += v;
  }
  ws[WS_PARTIAL + blk * 128 + d] = acc;
  if (d == 0) ws[WS_SVP + blk] = accsv;
}

// ---------------------------------------------------------------------------
// Kernel 2 (tiny, one block): fold partials, then
//   s[b] = Wk t[b] + bk*sv[b],  u[b] = Wq^T s[b],  c[b] = bq . s[b]
// ---------------------------------------------------------------------------
__global__ __launch_bounds__(128) void k2_project(const float* __restrict__ Wq,
                                                  const float* __restrict__ bq,
                                                  const float* __restrict__ Wk,
                                                  const float* __restrict__ bk,
                                                  float* __restrict__ ws) {
  __shared__ float t4[BATCH][128];
  __shared__ float s4[BATCH][128];
  __shared__ float sv[BATCH];
  const int d = threadIdx.x;
  for (int b = 0; b < BATCH; ++b) {
    float tv = 0.f;
    for (int s = 0; s < S1; ++s) tv += ws[WS_PARTIAL + (b * S1 + s) * 128 + d];
    t4[b][d] = tv;
  }
  if (d < BATCH) {
    float z = 0.f;
    for (int s = 0; s < S1; ++s) z += ws[WS_SVP + d * S1 + s];
    sv[d] = z;
  }
  __syncthreads();
  for (int b = 0; b < BATCH; ++b) {              // thread d = output row i of Wk
    float si = bk[d] * sv[b];
    for (int dd = 0; dd < 128; ++dd) si = fmaf(Wk[d * 128 + dd], t4[b][dd], si);
    s4[b][d] = si;
  }
  __syncthreads();
  for (int b = 0; b < BATCH; ++b) {              // u[d] = sum_i Wq[i,d]*s[i]
    float ud = 0.f;
    for (int i = 0; i < 128; ++i) ud = fmaf(Wq[i * 128 + d], s4[b][i], ud);
    ws[WS_U + b * 128 + d] = ud;
  }
  if (d < BATCH) {
    float cb = 0.f;
    for (int i = 0; i < 128; ++i) cb = fmaf(bq[i], s4[d][i], cb);
    ws[WS_C + d] = cb;
  }
}

// ---------------------------------------------------------------------------
// Kernel 3: fx[b,i] = scale * (h0[b,i,:] . u[b] + c[b])  via WMMA f32 16x16x4.
// Each wave owns a 16-row tile staged into LDS with async global->LDS loads
// (ASYNCcnt path) when the toolchain exposes the builtin; A = h0 tile,
// B = u replicated across all 16 columns with the same per-lane K placement
// as A, so every output column equals the per-row dot product
// (invariant to any within-chunk K permutation of the hardware layout).
// ---------------------------------------------------------------------------
__global__ __launch_bounds__(128) void k3_fx(const float* __restrict__ val,
                                             const float* __restrict__ id_embed,
                                             const float* __restrict__ ws,
                                             float* __restrict__ fx) {
  __shared__ float tile[W3][16 * 128];
  __shared__ float ulds[128];
  const int tid  = threadIdx.x;
  const int lane = tid & 31;
  const int w    = tid >> 5;
  const int g0   = blockIdx.x * (W3 * 16) + w * 16;   // first global row of tile
  const int bblk = (blockIdx.x * (W3 * 16)) / NROW;   // batch (block-uniform)

  ulds[tid] = ws[WS_U + bblk * 128 + tid];
  const float cb = ws[WS_C + bblk];

  // Stage 16x128 h0 tile (32 consecutive columns per step -> coalesced).
#ifdef HAVE_ASYNC_LDS
  for (int r = 0; r < 16; ++r) {
    const int gr = g0 + r;
    for (int t = 0; t < 4; ++t) {
      const int c = t * 32 + lane;
      const float* src = (c == 0) ? (val + gr)
                                  : (id_embed + (size_t)gr * 127 + (c - 1));
      __builtin_amdgcn_global_load_async_to_lds_b32(
          (gas_int*)src, (las_int*)&tile[w][r * 128 + c],
          /*imm_offset=*/0, /*cpol=*/0);
    }
  }
#if __has_builtin(__builtin_amdgcn_s_wait_asynccnt)
  __builtin_amdgcn_s_wait_asynccnt(0);
#else
  asm volatile("s_wait_asynccnt 0x0" ::: "memory");
#endif
#else
  for (int r = 0; r < 16; ++r) {
    const int gr = g0 + r;
    for (int t = 0; t < 4; ++t) {
      const int c = t * 32 + lane;
      const float x = (c == 0) ? val[gr] : id_embed[(size_t)gr * 127 + (c - 1)];
      tile[w][r * 128 + c] = x;
    }
  }
#endif
  __syncthreads();

  // f32 A layout (ISA 7.12.2): lanes 0-15 -> rows, {VGPR0,VGPR1} = {K,K+1};
  // lanes 16-31 -> same rows, {K+2,K+3}. Use the same placement for B(u).
  const int half = lane >> 4;
  const int m    = lane & 15;
  const v2f* arow = (const v2f*)&tile[w][m * 128];
  const v2f* urow = (const v2f*)ulds;

  v8f acc = {};
#pragma unroll
  for (int kc = 0; kc < 32; ++kc) {
    const v2f a  = arow[kc * 2 + half];
    const v2f bv = urow[kc * 2 + half];
    acc = __builtin_amdgcn_wmma_f32_16x16x4_f32(
        /*neg_a=*/false, a, /*neg_b=*/false, bv,
        /*c_mod=*/(short)0, acc, /*reuse_a=*/false, /*reuse_b=*/false);
  }

  const float scale = 0.08838834764831845f;  // 128^-0.5
  const float r0 = (acc[0] + cb) * scale;
  const float r1 = (acc[1] + cb) * scale;
  const float r2 = (acc[2] + cb) * scale;
  const float r3 = (acc[3] + cb) * scale;
  const float r4 = (acc[4] + cb) * scale;
  const float r5 = (acc[5] + cb) * scale;
  const float r6 = (acc[6] + cb) * scale;
  const float r7 = (acc[7] + cb) * scale;

  // C layout: VGPR v holds row v (lanes 0-15) / row v+8 (lanes 16-31);
  // all columns are identical, so lane 0 and lane 16 cover the 16 rows.
  if (lane == 0) {
    *(float4*)&fx[g0 + 0] = make_float4(r0, r1, r2, r3);
    *(float4*)&fx[g0 + 4] = make_float4(r4, r5, r6, r7);
  } else if (lane == 16) {
    *(float4*)&fx[g0 + 8]  = make_float4(r0, r1, r2, r3);
    *(float4*)&fx[g0 + 12] = make_float4(r4, r5, r6, r7);
  }
}

// ---------------------------------------------------------------------------
extern "C" void kernel_launch(void* const* d_in, const int* in_sizes, int n_in,
                              void* d_out, int out_size, void* d_ws, size_t ws_size,
                              hipStream_t stream) {
  const float* val      = (const float*)d_in[0];
  const float* id_embed = (const float*)d_in[1];
  const float* Wq       = (const float*)d_in[2];
  const float* bq       = (const float*)d_in[3];
  const float* Wk       = (const float*)d_in[4];
  const float* bk       = (const float*)d_in[5];
  float* ws = (float*)d_ws;
  float* fx = (float*)d_out;

  k1_reduce<<<BATCH * S1, 128, 0, stream>>>(val, id_embed, ws);
  k2_project<<<1, 128, 0, stream>>>(Wq, bq, Wk, bk, ws);
  k3_fx<<<(BATCH * NROW) / (W3 * 16), 128, 0, stream>>>(val, id_embed, ws, fx);
}